// AcceptRejectPooling2D_57526791963008
// MI455X (gfx1250) — compile-verified
//
#include <hip/hip_runtime.h>
#include <stdint.h>

// AcceptRejectPooling2D, inference mode, NHWC fp32.
//   x:   (64, 64, 64, 256)  -> out: (64, 32, 32, 256)
//   out = sum(relu(x)^2) / sum(relu(x)) per 2x2 window; 0 if window sums to 0.
//
// Purely bandwidth bound (~335 MB traffic, ~0.05 flop/byte) -> ~14 us floor at
// 23.3 TB/s. Optimization path: gfx1250 async global->LDS b128 copies
// (ASYNCcnt-tracked, double buffered), conflict-free ds_load_b128 reads,
// non-temporal b128 stores. WMMA intentionally unused: no contraction dim.

typedef float v4f __attribute__((ext_vector_type(4)));

#define B_DIM   64
#define H_DIM   64
#define W_DIM   64
#define C_DIM   256
#define HO_DIM  32
#define WO_DIM  32

#define THREADS 256
#define WOC     8                           // output columns per chunk
#define WIC     (2 * WOC)                   // input columns per chunk (16)
#define NCHUNK  (WO_DIM / WOC)              // 4 chunks per output row
#define V4_PER_BUF (2 * WIC * (C_DIM / 4))  // 2048 float4 = 32 KiB per buffer
#define UNITS_PER_THREAD (V4_PER_BUF / THREADS)  // 8 b128 copies / thread / chunk

// Generic -> AS(3) cast yields the raw LDS byte offset (addrspacecast strips
// the LDS aperture). Same pattern composable_kernel uses for global_load_lds.
__device__ __forceinline__ unsigned lds_offset_of(const void* p) {
  return (unsigned)(unsigned long long)(__attribute__((address_space(3))) const char*)p;
}

// GLOBAL_LOAD_ASYNC_TO_LDS_B128, GV mode: vdst = per-lane LDS byte offset,
// vaddr = per-lane 64-bit global address, saddr = off. Tracked by ASYNCcnt.
__device__ __forceinline__ void async_copy_b128(unsigned lds_byte_off, const void* gsrc) {
  asm volatile("global_load_async_to_lds_b128 %0, %1, off"
               :
               : "v"(lds_byte_off), "v"(gsrc)
               : "memory");
}

__device__ __forceinline__ v4f relu4(v4f v) {
  v.x = __builtin_fmaxf(v.x, 0.0f);
  v.y = __builtin_fmaxf(v.y, 0.0f);
  v.z = __builtin_fmaxf(v.z, 0.0f);
  v.w = __builtin_fmaxf(v.w, 0.0f);
  return v;
}

__device__ __forceinline__ float pool1(float ss, float s) {
  // s >= 0 after ReLU; zero window -> 0 (matches reference's safe divide).
  return s > 0.0f ? ss * __builtin_amdgcn_rcpf(s) : 0.0f;
}

__global__ __launch_bounds__(THREADS) void arpool_kernel(const float* __restrict__ x,
                                                         float* __restrict__ out) {
  // Double-buffered staging: [buf][row(2)][wi(16)][c/4(64)] float4
  __shared__ v4f ldsbuf[2][V4_PER_BUF];

  const int tid = threadIdx.x;
  const int bh  = blockIdx.x;        // 0 .. B*HO-1 (2048)
  const int b   = bh >> 5;
  const int ho  = bh & 31;

  const unsigned lds_base = lds_offset_of(&ldsbuf[0][0]);
  // Base of input row pair (hi = 2*ho), bytes.
  const char* xrow =
      (const char*)x + (size_t)((b * H_DIM + 2 * ho) * W_DIM) * C_DIM * 4u;

  auto issue_chunk = [&](int chunk, int buf) {
    const char* g0 = xrow + (size_t)(chunk * WIC) * C_DIM * 4u;  // row 0 of chunk
    const unsigned l0 = lds_base + (unsigned)buf * (V4_PER_BUF * 16);
#pragma unroll
    for (int k = 0; k < UNITS_PER_THREAD; ++k) {
      const int u   = k * THREADS + tid;  // 0..2047: wave covers 512B contiguous
      const int row = u >> 10;            // 1024 float4 per input row-span
      const int rem = u & 1023;
      async_copy_b128(l0 + (unsigned)u * 16,
                      g0 + (size_t)row * (W_DIM * C_DIM * 4u) + (size_t)rem * 16u);
    }
  };

  issue_chunk(0, 0);

  v4f* const outv = (v4f*)out;

  for (int ch = 0; ch < NCHUNK; ++ch) {
    const int cur = ch & 1;
    if (ch + 1 < NCHUNK) {
      // Prefetch next chunk into the other buffer (it was last read in the
      // previous iteration and is protected by the trailing barrier), then wait
      // for the current chunk: async loads complete in order, so <=8
      // outstanding means chunk `ch` has fully landed in LDS.
      issue_chunk(ch + 1, (ch + 1) & 1);
      asm volatile("s_wait_asynccnt 8" ::: "memory");
    } else {
      asm volatile("s_wait_asynccnt 0" ::: "memory");
    }
    __syncthreads();  // all waves' copy portions visible to all waves

    const v4f* Lb = &ldsbuf[cur][0];
#pragma unroll
    for (int j = 0; j < 2; ++j) {
      const int o  = j * THREADS + tid;   // 0..511 output float4s this chunk
      const int wo = o >> 6;              // 0..7
      const int cg = o & 63;              // channel group (4 floats)
      const int i00 = (2 * wo) * (C_DIM / 4) + cg;
      v4f a  = Lb[i00];                              // (r0, 2wo)
      v4f bb = Lb[i00 + (C_DIM / 4)];                // (r0, 2wo+1)
      v4f c  = Lb[i00 + WIC * (C_DIM / 4)];          // (r1, 2wo)
      v4f d  = Lb[i00 + (WIC + 1) * (C_DIM / 4)];    // (r1, 2wo+1)
      a = relu4(a); bb = relu4(bb); c = relu4(c); d = relu4(d);
      v4f s  = a + bb + c + d;
      v4f ss = a * a + bb * bb + c * c + d * d;
      v4f r;
      r.x = pool1(ss.x, s.x);
      r.y = pool1(ss.y, s.y);
      r.z = pool1(ss.z, s.z);
      r.w = pool1(ss.w, s.w);
      const size_t oidx =
          ((size_t)(b * HO_DIM + ho) * WO_DIM + (ch * WOC + wo)) * (C_DIM / 4) + cg;
      __builtin_nontemporal_store(r, outv + oidx);  // written once, never re-read
    }
    __syncthreads();  // buffer `cur` free for reuse next iteration
  }
}

extern "C" void kernel_launch(void* const* d_in, const int* in_sizes, int n_in,
                              void* d_out, int out_size, void* d_ws, size_t ws_size,
                              hipStream_t stream) {
  (void)in_sizes; (void)n_in; (void)d_ws; (void)ws_size; (void)out_size;
  const float* x = (const float*)d_in[0];
  float* out = (float*)d_out;
  arpool_kernel<<<dim3(B_DIM * HO_DIM), dim3(THREADS), 0, stream>>>(x, out);
}